// LightGCN_35253091566188
// MI455X (gfx1250) — compile-verified
//
#include <hip/hip_runtime.h>
#include <cstdint>
#include <cstddef>

// ---------------------------------------------------------------------------
// LightGCN propagation for MI455X (gfx1250, wave32).
//   acc = x;  for 3 layers: x = SpMM(A, x); acc += x
// - one wave32 per edge: lane l owns dims {2l, 2l+1}  (32 x float2 = 64 dims)
// - edge triples streamed into LDS by the Tensor Data Mover, double-buffered
//   across grid-stride tiles (issue next tile, s_wait_tensorcnt<=3 for current)
// - TDM loads marked non-temporal: edge lists are stream-once, keep L2 for
//   the x-table + y-accumulator working set (~153 MB vs 192 MB L2)
// - scatter via native fp32 global atomics (64 per edge, the HW minimum)
// ---------------------------------------------------------------------------

#define TILE        1024      // edges per LDS tile; 2 buffers * 12 KB = 24 KB LDS
#define SPMM_BLOCK  256       // 8 waves
#define SPMM_MAXGRID 2048     // persistent blocks (enough to fill all WGPs)
#define CPY_BLOCK   256

typedef uint32_t u32x4 __attribute__((ext_vector_type(4)));
typedef uint32_t u32x8 __attribute__((ext_vector_type(8)));
typedef float    f32x4 __attribute__((ext_vector_type(4)));

// ---- Tensor Data Mover: 1-D tile load, 4-byte elements --------------------
// D# per CDNA5 ISA ch.8. tensor_dim0 = remaining elements => hardware
// zero-fills LDS for the out-of-bounds tail of the last tile.
__device__ __forceinline__ void tdm_load_1d(uint32_t lds_byte_addr,
                                            const void* gsrc,
                                            uint32_t remaining_elems,
                                            uint32_t tile_elems)
{
    uint64_t ga = (uint64_t)(uintptr_t)gsrc;

    u32x4 g0;
    g0[0] = 0x1u;                                            // count=1 (valid), user mode
    g0[1] = lds_byte_addr;                                   // lds_addr [63:32]
    g0[2] = (uint32_t)ga;                                    // global_addr[31:0]
    g0[3] = ((uint32_t)(ga >> 32) & 0x01FFFFFFu)             // global_addr[56:32]
          | (2u << 30);                                      // type = 2 ("image")

    u32x8 g1;
    g1[0] = (2u << 16);                                      // data_size = 2 -> 4 bytes
    g1[1] = (remaining_elems & 0xFFFFu) << 16;               // tensor_dim0[15:0]  @ bits 63:48
    g1[2] = (remaining_elems >> 16) | (1u << 16);            // tensor_dim0[31:16], tensor_dim1 = 1
    g1[3] = (tile_elems & 0xFFFFu) << 16;                    // tile_dim0 @ bits 127:112
    g1[4] = 0u;                                              // tile_dim1 = 0, tile_dim2 = 0
    g1[5] = remaining_elems;                                 // tensor_dim0_stride lo (unused in 1D)
    g1[6] = 0u;
    g1[7] = 0u;

    // Non-temporal: edge stream is read exactly once per layer.
    asm volatile("tensor_load_to_lds %0, %1 th:TH_LOAD_NT"
                 :
                 : "s"(g0), "s"(g1)
                 : "memory");
}

__device__ __forceinline__ uint32_t lds_addr_of(const void* p)
{
    // Flat pointer to a __shared__ object: low 32 bits are the LDS byte offset.
    return (uint32_t)(uintptr_t)p;
}

// ---------------------------------------------------------------------------
__global__ void __launch_bounds__(SPMM_BLOCK)
k_spmm(const int* __restrict__ rows, const int* __restrict__ cols,
       const float* __restrict__ vals, const float* __restrict__ x,
       float* __restrict__ y, int nEdges)
{
    __shared__ int   s_rows[2][TILE];
    __shared__ int   s_cols[2][TILE];
    __shared__ float s_vals[2][TILE];

    const int lane    = threadIdx.x & 31;
    const int wave    = threadIdx.x >> 5;
    const int nwaves  = SPMM_BLOCK / 32;     // 8
    const int perWave = TILE / nwaves;       // 128 edges per wave per tile

    const int ntiles = (nEdges + TILE - 1) / TILE;
    const int G      = gridDim.x;

    // Pipeline prologue: DMA this block's first tile into buffer 0.
    if (wave == 0) {
        const int base = blockIdx.x * TILE;
        const uint32_t rem = (uint32_t)(nEdges - base);
        tdm_load_1d(lds_addr_of(&s_rows[0][0]), rows + base, rem, TILE);
        tdm_load_1d(lds_addr_of(&s_cols[0][0]), cols + base, rem, TILE);
        tdm_load_1d(lds_addr_of(&s_vals[0][0]), vals + base, rem, TILE);
    }

    int buf = 0;
    for (int t = blockIdx.x; t < ntiles; t += G, buf ^= 1) {
        const int base = t * TILE;
        int cnt = nEdges - base;
        if (cnt > TILE) cnt = TILE;

        if (wave == 0) {
            const int nt = t + G;
            if (nt < ntiles) {
                // Issue next tile into the other buffer, then wait for the
                // current tile's 3 loads (TENSORcnt completes in order).
                const int nb = buf ^ 1;
                const int nbase = nt * TILE;
                const uint32_t rem = (uint32_t)(nEdges - nbase);
                tdm_load_1d(lds_addr_of(&s_rows[nb][0]), rows + nbase, rem, TILE);
                tdm_load_1d(lds_addr_of(&s_cols[nb][0]), cols + nbase, rem, TILE);
                tdm_load_1d(lds_addr_of(&s_vals[nb][0]), vals + nbase, rem, TILE);
                __builtin_amdgcn_s_wait_tensorcnt(3);
            } else {
                __builtin_amdgcn_s_wait_tensorcnt(0);
            }
        }
        __syncthreads();                      // current buffer visible to all waves

        const int eBeg = wave * perWave;
        int eEnd = eBeg + perWave;
        if (eEnd > cnt) eEnd = cnt;

        for (int e = eBeg; e < eEnd; ++e) {
            const int   c = s_cols[buf][e];
            const int   r = s_rows[buf][e];
            const float v = s_vals[buf][e];

            if (e + 1 < eEnd) {               // warm caches for next gather row
                __builtin_prefetch(x + (((size_t)s_cols[buf][e + 1]) << 6) + (lane << 1), 0, 3);
            }

            const float2 xv = *(const float2*)(x + (((size_t)c) << 6) + (lane << 1));
            float* yp = y + (((size_t)r) << 6) + (lane << 1);
            unsafeAtomicAdd(yp,     v * xv.x);
            unsafeAtomicAdd(yp + 1, v * xv.y);
        }
        __syncthreads();                      // all waves done with buf before it is re-DMA'd
    }
}

// ---- fused elementwise passes (f32x4 vectorized, b128 traffic) ------------
// init: acc = x0 = concat(u, i);  y0 = 0
__global__ void __launch_bounds__(CPY_BLOCK)
k_concat_zero(const f32x4* __restrict__ u, const f32x4* __restrict__ it,
              f32x4* __restrict__ acc, f32x4* __restrict__ x0,
              f32x4* __restrict__ y0, int nU4, int nTot4)
{
    int i = blockIdx.x * blockDim.x + threadIdx.x;
    if (i >= nTot4) return;
    f32x4 v = (i < nU4) ? u[i] : it[i - nU4];
    __builtin_nontemporal_store(v, &acc[i]);  // acc not re-read until k_acc_zero
    x0[i] = v;                                // hot next kernel: keep in L2
    y0[i] = (f32x4)(0.f);                     // atomic target: keep in L2
}

// acc += y; zero the retired x buffer (it becomes the next layer's y)
__global__ void __launch_bounds__(CPY_BLOCK)
k_acc_zero(f32x4* __restrict__ acc, const f32x4* __restrict__ y,
           f32x4* __restrict__ xold, int n4, int doZero)
{
    int i = blockIdx.x * blockDim.x + threadIdx.x;
    if (i >= n4) return;
    f32x4 a = __builtin_nontemporal_load(&acc[i]);   // acc is stream-through
    f32x4 b = y[i];                                  // y stays hot (next layer's x)
    a = a + b;
    __builtin_nontemporal_store(a, &acc[i]);
    if (doZero) xold[i] = (f32x4)(0.f);
}

// ---------------------------------------------------------------------------
extern "C" void kernel_launch(void* const* d_in, const int* in_sizes, int n_in,
                              void* d_out, int out_size, void* d_ws, size_t ws_size,
                              hipStream_t stream)
{
    const int*   rows = (const int*)  d_in[0];
    const int*   cols = (const int*)  d_in[1];
    const float* vals = (const float*)d_in[2];
    const float* u    = (const float*)d_in[3];
    const float* it   = (const float*)d_in[4];

    const int E    = in_sizes[0];
    const int nU   = in_sizes[3];        // USER * 64
    const int nI   = in_sizes[4];        // ITEM * 64
    const int nTot = nU + nI;            // N * 64
    const int n4   = nTot / 4;

    float* acc  = (float*)d_out;
    float* bufA = (float*)d_ws;          // x / y ping-pong, 2 * N * 64 floats
    float* bufB = bufA + (size_t)nTot;

    const int cpyGrid = (n4 + CPY_BLOCK - 1) / CPY_BLOCK;
    k_concat_zero<<<cpyGrid, CPY_BLOCK, 0, stream>>>(
        (const f32x4*)u, (const f32x4*)it,
        (f32x4*)acc, (f32x4*)bufA, (f32x4*)bufB, nU / 4, n4);

    float* x  = bufA;
    float* yb = bufB;
    const int ntiles = (E + TILE - 1) / TILE;
    const int G = (ntiles < SPMM_MAXGRID) ? ntiles : SPMM_MAXGRID;

    for (int l = 0; l < 3; ++l) {
        k_spmm<<<G, SPMM_BLOCK, 0, stream>>>(rows, cols, vals, x, yb, E);
        // acc += y; zero retired x so it can serve as the next layer's y
        k_acc_zero<<<cpyGrid, CPY_BLOCK, 0, stream>>>(
            (f32x4*)acc, (const f32x4*)yb, (f32x4*)x, n4, (l < 2) ? 1 : 0);
        float* tmp = x; x = yb; yb = tmp;
    }
}